// BeliefPropagationMul_61564061221584
// MI455X (gfx1250) — compile-verified
//
#include <hip/hip_runtime.h>

// ---------------------------------------------------------------------------
// BeliefPropagation on junction tree — MI455X (gfx1250) implementation.
// Memory-bound: ~1.5 GB total traffic => ~65us at 23.3 TB/s. The selection-
// matrix einsums are group-sums / upsamples; the theta0 reduction+pool runs
// on the matrix pipe with V_WMMA_F32_16X16X4_F32 (A = faithful one-hot tile of
// m_x1_fwd^T); everything else is float4 streaming sized for latency coverage.
// ---------------------------------------------------------------------------

typedef float v2f __attribute__((ext_vector_type(2)));
typedef float v8f __attribute__((ext_vector_type(8)));

#define N_TH0 67108864   // 256*1024*256
#define N_TH1 33554432   // 256*256*512
#define N_TH2 33554432   // 256*256*512

// workspace layout (float offsets)
#define WS_PART   0u        // 32 * 65536 msg01 partials (per x0-chunk)
#define WS_T1S0P  2097152u  // 4 * 131072 t1s0 partials (per c1-chunk)
#define WS_MSG01  2621440u  // 256*256
#define WS_T1S0   2686976u  // 256*512  theta1.sum(0)   [x2][x3f]
#define WS_T1S2   2818048u  // 256*256  theta1.sum(2)   [c1][x2]
#define WS_T2S2   2883584u  // 256*256  theta2.sum(2)   [x2][x3c]
#define WS_MSG12  2949120u  // 256*256  [x2][x3c]
#define WS_A      3014656u  // 256*256  a = s2 - msg12 = t2s2 + 511*msg12
#define WS_RS21   3080192u  // 256      rowsum21[x2] = 2*sum_x3c a
#define WS_B      3080448u  // 256*256  b = t1s2 + 511*msg01 + rs21[x2]

// ---------------------------------------------------------------------------
// k1: msg01 partials via WMMA f32 16x16x4.
//   msg01[c1,x2] = sum_{x0} sum_{d<4} theta0[x0, 4*c1+d, x2]
// Per wave: one 16x16 (c1,x2) tile, 8 x0 values, 16 j-steps (j = local coarse
// row). A-tile of m_x1_fwd^T for fine block f0=4*(c1base+j) is e_j (x) ones(4):
//   A[m][k] = (m == j).  B[k][n] = theta0[x0, f0+k, x2base+n].
// B VGPR layout (4x16 f32): V0 = {K0 lanes0-15, K1 lanes16-31}, V1 = {K2, K3}.
// ---------------------------------------------------------------------------
__global__ __launch_bounds__(32) void k1_msg01_wmma(
    const float* __restrict__ th0, float* __restrict__ part) {
  const int bid   = blockIdx.x;
  const int tile  = bid >> 5;          // 0..255
  const int chunk = bid & 31;          // x0 chunk (8 x0 each)
  const int c1base = (tile >> 4) << 4;
  const int x2base = (tile & 15) << 4;
  const int lane = threadIdx.x;
  const int col  = lane & 15;          // N within tile
  const int kb   = lane >> 4;          // 0: rows K0/K2, 1: rows K1/K3

  v8f acc = {0.f, 0.f, 0.f, 0.f, 0.f, 0.f, 0.f, 0.f};

  for (int x0 = chunk * 8; x0 < chunk * 8 + 8; ++x0) {
    const float* rp = th0 + (size_t)x0 * 262144
                          + (size_t)(4 * c1base + kb) * 256 + x2base + col;
#pragma unroll
    for (int j = 0; j < 16; ++j) {
      const float bx = rp[j * 1024];        // row f0 + kb
      const float by = rp[j * 1024 + 512];  // row f0 + kb + 2
      v2f B = {bx, by};
      const float av = (col == j) ? 1.0f : 0.0f;  // one-hot row of m_x1_fwd^T
      v2f A = {av, av};
      acc = __builtin_amdgcn_wmma_f32_16x16x4_f32(
          false, A, false, B, (short)0, acc, false, false);
    }
  }

  // C/D layout: VGPR r -> (M=r, N=lane) lanes0-15 ; (M=r+8, N=lane-16) lanes16-31
  float* p = part + (size_t)chunk * 65536;
#pragma unroll
  for (int r = 0; r < 8; ++r) {
    const int row = c1base + r + (kb ? 8 : 0);
    p[row * 256 + x2base + col] = acc[r];
  }
}

// k1b: deterministic reduction of the 32 x0-chunk partials -> msg01
__global__ __launch_bounds__(256) void k1b_reduce(
    const float* __restrict__ part, float* __restrict__ msg01) {
  const int i = blockIdx.x * 256 + threadIdx.x;
  float s = 0.f;
#pragma unroll
  for (int c = 0; c < 32; ++c) s += part[c * 65536 + i];
  msg01[i] = s;
}

// ---------------------------------------------------------------------------
// k2: theta1 pass (float4). Block = (x2, 64-c1 chunk), 128 threads.
//  - t1s0 partial per chunk (register float4 accumulate over 64 c1)
//  - t1s2 exact (each c1 row fully inside the block: wave shfl + 4-slot LDS)
// ---------------------------------------------------------------------------
__global__ __launch_bounds__(128) void k2_theta1_sums(
    const float* __restrict__ th1, float* __restrict__ t1s0p,
    float* __restrict__ t1s2) {
  __shared__ float lds_part[64 * 4];  // [c1_local][wave]
  const int x2 = blockIdx.x & 255;
  const int chunk = blockIdx.x >> 8;  // 0..3
  const int tid = threadIdx.x;
  const int w = tid >> 5, lane = tid & 31;
  const float4* th1v = (const float4*)th1;
  float4 acc = {0.f, 0.f, 0.f, 0.f};
  for (int i = 0; i < 64; ++i) {
    const int c1 = chunk * 64 + i;
    const float4 v = th1v[(size_t)c1 * 32768 + (size_t)x2 * 128 + tid];
    acc.x += v.x; acc.y += v.y; acc.z += v.z; acc.w += v.w;
    float r = v.x + v.y + v.z + v.w;
    for (int o = 16; o > 0; o >>= 1) r += __shfl_xor(r, o, 32);
    if (lane == 0) lds_part[i * 4 + w] = r;
  }
  ((float4*)t1s0p)[(size_t)chunk * 32768 + (size_t)x2 * 128 + tid] = acc;
  __syncthreads();
  if (tid < 64) {
    const float s = lds_part[tid * 4] + lds_part[tid * 4 + 1]
                  + lds_part[tid * 4 + 2] + lds_part[tid * 4 + 3];
    t1s2[(chunk * 64 + tid) * 256 + x2] = s;  // [c1][x2]
  }
}

// k2b: reduce the 4 c1-chunk partials -> t1s0 [x2][x3f]
__global__ __launch_bounds__(256) void k2b_reduce(
    const float* __restrict__ t1s0p, float* __restrict__ t1s0) {
  const int i = blockIdx.x * 256 + threadIdx.x;
  t1s0[i] = t1s0p[i] + t1s0p[131072 + i] + t1s0p[262144 + i]
          + t1s0p[393216 + i];
}

// k3: t2s2[x2*256+x3c] = sum over x4 (contiguous 512) — wave per row, float4.
__global__ __launch_bounds__(256) void k3_theta2_rowsum(
    const float* __restrict__ th2, float* __restrict__ t2s2) {
  const int row = blockIdx.x * 8 + (threadIdx.x >> 5);
  const int lane = threadIdx.x & 31;
  const float4* base = (const float4*)(th2 + (size_t)row * 512);
  float s = 0.f;
#pragma unroll
  for (int k = 0; k < 4; ++k) {
    const float4 v = base[lane + 32 * k];
    s += v.x + v.y + v.z + v.w;
  }
  for (int o = 16; o > 0; o >>= 1) s += __shfl_xor(s, o, 32);
  if (lane == 0) t2s2[row] = s;
}

// kM1: per-x2 message algebra: csum01, msg12, a, rowsum21.
__global__ __launch_bounds__(256) void kM1(
    const float* __restrict__ msg01, const float* __restrict__ t1s0,
    const float* __restrict__ t2s2, float* __restrict__ msg12,
    float* __restrict__ amat, float* __restrict__ rs21) {
  __shared__ float red[256];
  const int x2 = blockIdx.x, t = threadIdx.x;
  red[t] = msg01[t * 256 + x2];
  __syncthreads();
  for (int s = 128; s > 0; s >>= 1) {
    if (t < s) red[t] += red[t + s];
    __syncthreads();
  }
  const float csum = red[0];
  __syncthreads();
  // s1 = t1s0 + csum ; msg12 = pairsum(s1)
  const float m12 = t1s0[x2 * 512 + 2 * t] + t1s0[x2 * 512 + 2 * t + 1]
                    + 2.0f * csum;
  msg12[x2 * 256 + t] = m12;
  // a = (t2s2 + 512*msg12) - msg12
  const float av = t2s2[x2 * 256 + t] + 511.0f * m12;
  amat[x2 * 256 + t] = av;
  red[t] = av;
  __syncthreads();
  for (int s = 128; s > 0; s >>= 1) {
    if (t < s) red[t] += red[t + s];
    __syncthreads();
  }
  if (t == 0) rs21[x2] = 2.0f * red[0];  // sum_x3f msg21[x2,:]
}

// kM2: b[c1,x2] = t1s2 + 511*msg01 + rowsum21[x2]
__global__ __launch_bounds__(256) void kM2(
    const float* __restrict__ t1s2, const float* __restrict__ msg01,
    const float* __restrict__ rs21, float* __restrict__ bmat) {
  const int c1 = blockIdx.x, t = threadIdx.x;
  const int idx = c1 * 256 + t;
  bmat[idx] = t1s2[idx] + 511.0f * msg01[idx] + rs21[t];
}

// k4: out1 = theta1 + msg01[c1,x2] + a[x2, x3f>>1]   (float4 streaming)
__global__ __launch_bounds__(128) void k4_update1(
    const float* __restrict__ th1, const float* __restrict__ msg01,
    const float* __restrict__ amat, float* __restrict__ out1) {
  const int row = blockIdx.x;       // c1*256 + x2
  const int x2 = row & 255;
  const int t = threadIdx.x;
  const size_t base = (size_t)row * 512 + 4 * t;
  float4 v = *(const float4*)(th1 + base);
  const float m = msg01[row];
  const float a0 = amat[x2 * 256 + 2 * t];
  const float a1 = amat[x2 * 256 + 2 * t + 1];
  v.x += m + a0; v.y += m + a0; v.z += m + a1; v.w += m + a1;
  *(float4*)(out1 + base) = v;
}

// k5: out2 = theta2 + msg12[x2,x3c]   (float4 streaming)
__global__ __launch_bounds__(128) void k5_update2(
    const float* __restrict__ th2, const float* __restrict__ msg12,
    float* __restrict__ out2) {
  const int row = blockIdx.x;       // x2*256 + x3c
  const int t = threadIdx.x;
  const size_t base = (size_t)row * 512 + 4 * t;
  float4 v = *(const float4*)(th2 + base);
  const float m = msg12[row];
  v.x += m; v.y += m; v.z += m; v.w += m;
  *(float4*)(out2 + base) = v;
}

// k6: out0 = theta0 + b[f>>2, x2]  (float4; b row float4 cached in register)
__global__ __launch_bounds__(256) void k6_update0(
    const float* __restrict__ th0, const float* __restrict__ bmat,
    float* __restrict__ out0) {
  const int f = blockIdx.x >> 3;
  const int chunk = blockIdx.x & 7;
  const int t = threadIdx.x;
  const int x2v = t & 63;    // float4 index within the 256-wide x2 row
  const int xo = t >> 6;     // 0..3 : x0 sub-offset
  const float4 bb = ((const float4*)(bmat + (f >> 2) * 256))[x2v];
  const float4* th0v = (const float4*)th0;
  float4* out0v = (float4*)out0;
#pragma unroll
  for (int it = 0; it < 8; ++it) {
    const int x0 = chunk * 32 + it * 4 + xo;
    const size_t idx = (size_t)x0 * 65536 + (size_t)f * 64 + x2v;
    float4 v = th0v[idx];
    v.x += bb.x; v.y += bb.y; v.z += bb.z; v.w += bb.w;
    out0v[idx] = v;
  }
}

extern "C" void kernel_launch(void* const* d_in, const int* in_sizes, int n_in,
                              void* d_out, int out_size, void* d_ws,
                              size_t ws_size, hipStream_t stream) {
  (void)in_sizes; (void)n_in; (void)out_size; (void)ws_size;
  const float* th0 = (const float*)d_in[0];
  const float* th1 = (const float*)d_in[1];
  const float* th2 = (const float*)d_in[2];
  // d_in[3..6] are the deterministic selection matrices; their structure
  // (M[f, f//r] = 1) is folded into the pooled/upsampled indexing above.
  float* ws = (float*)d_ws;
  float* out0 = (float*)d_out;
  float* out1 = out0 + N_TH0;
  float* out2 = out1 + N_TH1;

  // phase A: reductions
  k1_msg01_wmma<<<8192, 32, 0, stream>>>(th0, ws + WS_PART);
  k1b_reduce<<<256, 256, 0, stream>>>(ws + WS_PART, ws + WS_MSG01);
  k2_theta1_sums<<<1024, 128, 0, stream>>>(th1, ws + WS_T1S0P, ws + WS_T1S2);
  k2b_reduce<<<512, 256, 0, stream>>>(ws + WS_T1S0P, ws + WS_T1S0);
  k3_theta2_rowsum<<<8192, 256, 0, stream>>>(th2, ws + WS_T2S2);
  // phase B: tiny message algebra
  kM1<<<256, 256, 0, stream>>>(ws + WS_MSG01, ws + WS_T1S0, ws + WS_T2S2,
                               ws + WS_MSG12, ws + WS_A, ws + WS_RS21);
  kM2<<<256, 256, 0, stream>>>(ws + WS_T1S2, ws + WS_MSG01, ws + WS_RS21,
                               ws + WS_B);
  // phase C: streaming updates
  k4_update1<<<65536, 128, 0, stream>>>(th1, ws + WS_MSG01, ws + WS_A, out1);
  k5_update2<<<65536, 128, 0, stream>>>(th2, ws + WS_MSG12, out2);
  k6_update0<<<8192, 256, 0, stream>>>(th0, ws + WS_B, out0);
}